// MMGCNLayer_43516608643383
// MI455X (gfx1250) — compile-verified
//
#include <hip/hip_runtime.h>

typedef __attribute__((ext_vector_type(2))) float v2f;
typedef __attribute__((ext_vector_type(8))) float v8f;

#define DFEAT 128

// ---------------------------------------------------------------------------
// h[M,128] = x[M,128] @ W[128,128]   (fp32, V_WMMA_F32_16X16X4_F32)
// One wave computes one 16x16 tile; block = 8 waves = one 16-row strip x 128 cols.
// A 16x4 f32 layout: lanes 0-15 hold M=lane, VGPR0=K{0|2}, VGPR1=K{1|3}
//                    (lane>=16 -> K base +2).
// D 16x16 f32 layout: VGPR j -> M = j + (lane>=16 ? 8 : 0), N = lane&15.
// ---------------------------------------------------------------------------
__global__ void gemm_wmma_f32(const float* __restrict__ x,
                              const float* __restrict__ w,
                              float* __restrict__ h) {
  const int lane = threadIdx.x & 31;
  const int wid  = threadIdx.x >> 5;      // 0..7 -> N tile
  const int m0   = blockIdx.x * 16;
  const int n0   = wid * 16;
  const int l16  = lane & 15;
  const int kh   = (lane >> 4) << 1;      // 0 or 2

  const float* arow = x + (m0 + l16) * DFEAT;   // this lane's A row
  const float* bcol = w + n0 + l16;             // this lane's B column

  v8f c = {};
#pragma unroll
  for (int k0 = 0; k0 < DFEAT; k0 += 4) {
    v2f a = *(const v2f*)(arow + k0 + kh);      // K=k0+kh, k0+kh+1 (contig 8B)
    v2f b;
    b.x = bcol[(k0 + kh    ) * DFEAT];          // B row K=k0+kh
    b.y = bcol[(k0 + kh + 1) * DFEAT];          // B row K=k0+kh+1
    // 8 args: (neg_a, A, neg_b, B, c_mod, C, reuse_a, reuse_b)
    c = __builtin_amdgcn_wmma_f32_16x16x4_f32(false, a, false, b,
                                              (short)0, c, false, false);
  }

  float* hcol = h + n0 + l16;
  const int rbase = m0 + ((lane >> 4) << 3);    // +8 rows for lanes 16-31
#pragma unroll
  for (int j = 0; j < 8; ++j) {
    hcol[(rbase + j) * DFEAT] = c[j];
  }
}

// ---------------------------------------------------------------------------
// out[i] = bias[i % 128]   (fresh every launch; atomics accumulate on top)
// ---------------------------------------------------------------------------
__global__ void init_out_bias(const float* __restrict__ bias,
                              float* __restrict__ out, int n) {
  int i = blockIdx.x * blockDim.x + threadIdx.x;
  if (i < n) out[i] = bias[i & (DFEAT - 1)];
}

// ---------------------------------------------------------------------------
// COO SpMM scatter: out[row[e], :] += val[e] * h[col[e], :]
// One wave per 4 edges; lane handles 4 features via float4 gather + 4 fp32
// atomics (global_atomic_add_f32, resolved in L2: out is 5.12 MB, L2-resident).
// ---------------------------------------------------------------------------
__global__ void spmm_scatter(const float* __restrict__ h,
                             const float* __restrict__ val,
                             const int* __restrict__ rowi,
                             const int* __restrict__ coli,
                             float* __restrict__ out, int E) {
  const int lane = threadIdx.x & 31;
  const int wid  = threadIdx.x >> 5;          // 0..7
  const int d0   = lane << 2;                 // feature base, 0..124
  int e = (blockIdx.x * 8 + wid) * 4;

#pragma unroll
  for (int i = 0; i < 4; ++i, ++e) {
    if (e >= E) return;
    const int   r = rowi[e];
    const int   cidx = coli[e];
    const float v = val[e];
    const float4 hv = *(const float4*)(h + cidx * DFEAT + d0);
    float* o = out + r * DFEAT + d0;
    unsafeAtomicAdd(o + 0, v * hv.x);
    unsafeAtomicAdd(o + 1, v * hv.y);
    unsafeAtomicAdd(o + 2, v * hv.z);
    unsafeAtomicAdd(o + 3, v * hv.w);
  }
}

// ---------------------------------------------------------------------------
extern "C" void kernel_launch(void* const* d_in, const int* in_sizes, int n_in,
                              void* d_out, int out_size, void* d_ws, size_t ws_size,
                              hipStream_t stream) {
  const float* x       = (const float*)d_in[0];   // [N,128]
  const float* adj_val = (const float*)d_in[1];   // [E]
  const float* weight  = (const float*)d_in[2];   // [128,128]
  const float* bias    = (const float*)d_in[3];   // [128]
  const int*   adj_row = (const int*)d_in[4];     // [E]
  const int*   adj_col = (const int*)d_in[5];     // [E]

  const int N = in_sizes[0] / DFEAT;              // 10000
  const int E = in_sizes[1];                      // 640000

  float* h   = (float*)d_ws;                      // [N,128] scratch (5.12 MB)
  float* out = (float*)d_out;                     // [N,128]

  // 1) h = x @ W  (WMMA fp32)
  gemm_wmma_f32<<<N / 16, 256, 0, stream>>>(x, weight, h);

  // 2) out = bias (broadcast)
  init_out_bias<<<(N * DFEAT + 255) / 256, 256, 0, stream>>>(bias, out, N * DFEAT);

  // 3) out[row] += val * h[col]
  const int waves  = (E + 3) / 4;
  const int blocks = (waves + 7) / 8;
  spmm_scatter<<<blocks, 256, 0, stream>>>(h, adj_val, adj_row, adj_col, out, E);
}